// NODE_REN_54305566490953
// MI455X (gfx1250) — compile-verified
//
#include <hip/hip_runtime.h>
#include <math.h>

#define NXD 256
#define NQD 256
#define NUD 128
#define NBD 8192
#define NPQ 512
#define EPSV 1e-3f
#define AUGW 768   // [P | Y | RHS] augmented width

typedef __attribute__((ext_vector_type(2))) float v2f;
typedef __attribute__((ext_vector_type(8))) float v8f;

// ---------------- K1: H = X @ X^T + eps*I  (512x512) ----------------
__global__ __launch_bounds__(256) void k_H(const float* __restrict__ X,
                                           float* __restrict__ H) {
    int idx = blockIdx.x * blockDim.x + threadIdx.x;   // 0 .. 512*512-1
    int i = idx >> 9;
    int j = idx & 511;
    const float* xi = X + i * NPQ;
    const float* xj = X + j * NPQ;
    float s = 0.f;
    for (int k = 0; k < NPQ; ++k) s = fmaf(xi[k], xj[k], s);
    if (i == j) s += EPSV;
    H[idx] = s;
}

// ---- K2: build Paug=[P | Y | -H2-Chi], D11, C1 (lam folded in) ----
__global__ __launch_bounds__(256) void k_prep(const float* __restrict__ H,
                                              const float* __restrict__ Pstar,
                                              const float* __restrict__ Y1,
                                              const float* __restrict__ Chi,
                                              float* __restrict__ Paug,
                                              float* __restrict__ D11,
                                              float* __restrict__ C1) {
    int idx = blockIdx.x * blockDim.x + threadIdx.x;   // 0 .. 256*256-1
    int i = idx >> 8;
    int j = idx & 255;
    // P = 0.5 * Pstar @ Pstar^T + eps*I
    const float* pi = Pstar + i * NXD;
    const float* pj = Pstar + j * NXD;
    float p = 0.f;
    for (int k = 0; k < NXD; ++k) p = fmaf(pi[k], pj[k], p);
    p *= 0.5f;
    if (i == j) p += EPSV;
    Paug[i * AUGW + j] = p;
    // Y = -0.5*(H1 + Y1 - Y1^T)   (ALPHA == 0)
    Paug[i * AUGW + 256 + j] =
        -0.5f * (H[i * NPQ + j] + Y1[i * NXD + j] - Y1[j * NXD + i]);
    // RHS = -H2 - Chi
    Paug[i * AUGW + 512 + j] = -H[i * NPQ + 256 + j] - Chi[i * NQD + j];
    // lam[i] = 0.5*H4[i][i];  D11 = -tril(H4,-1)/lam;  C1 = Chi^T/lam
    float lam_i = 0.5f * H[(256 + i) * NPQ + 256 + i];
    float h4ij  = H[(256 + i) * NPQ + 256 + j];
    D11[i * NQD + j] = (j < i) ? (-h4ij / lam_i) : 0.f;
    C1[i * NXD + j]  = Chi[j * NQD + i] / lam_i;
}

// ---- K3: Gauss-Jordan on [P|Y|RHS] -> [I|A|B1]; single workgroup ----
__global__ __launch_bounds__(1024) void k_gj(float* __restrict__ M) {
    __shared__ float piv[AUGW];
    __shared__ float fcol[256];
    const int tid = threadIdx.x;
    for (int k = 0; k < 256; ++k) {
        float invp = 1.0f / M[k * AUGW + k];           // broadcast read
        if (tid < 256) fcol[tid] = (tid == k) ? 0.f : M[tid * AUGW + k];
        __syncthreads();
        for (int c = tid; c < AUGW; c += 1024) {
            float v = M[k * AUGW + c] * invp;
            piv[c] = v;
            M[k * AUGW + c] = v;
        }
        __syncthreads();
        for (int idx = tid; idx < 256 * AUGW; idx += 1024) {
            int r = idx / AUGW;
            if (r != k) {
                int c = idx - r * AUGW;
                M[idx] = fmaf(-fcol[r], piv[c], M[idx]);
            }
        }
        __syncthreads();
    }
}

// ---- K4: base = xi @ C1^T + u @ D12^T + bv   (8192 x 256), WMMA f32 ----
__global__ __launch_bounds__(256) void k_base(const float* __restrict__ xi,
                                              const float* __restrict__ u,
                                              const float* __restrict__ C1,
                                              const float* __restrict__ D12,
                                              const float* __restrict__ bv,
                                              float* __restrict__ base) {
#if defined(__gfx1250__) && __has_builtin(__builtin_amdgcn_wmma_f32_16x16x4_f32)
    const int lane = threadIdx.x & 31;
    const int wave = threadIdx.x >> 5;
    const int tile = blockIdx.x * 8 + wave;            // 8192 tiles total
    const int tm = tile >> 4;                          // 512 M-tiles
    const int tn = tile & 15;                          // 16 N-tiles
    const int l15 = lane & 15;
    const int kh  = (lane >> 4) * 2;                   // K sub-offset 0 / 2
    const float* aX = xi  + (tm * 16 + l15) * NXD + kh;
    const float* aU = u   + (tm * 16 + l15) * NUD + kh;
    const float* bC = C1  + (tn * 16 + l15) * NXD + kh;
    const float* bD = D12 + (tn * 16 + l15) * NUD + kh;
    v8f acc = {};
    for (int k0 = 0; k0 < NXD; k0 += 4) {
        v2f a = *(const v2f*)(aX + k0);
        v2f b = *(const v2f*)(bC + k0);
        acc = __builtin_amdgcn_wmma_f32_16x16x4_f32(false, a, false, b,
                                                    (short)0, acc, false, false);
    }
    for (int k0 = 0; k0 < NUD; k0 += 4) {
        v2f a = *(const v2f*)(aU + k0);
        v2f b = *(const v2f*)(bD + k0);
        acc = __builtin_amdgcn_wmma_f32_16x16x4_f32(false, a, false, b,
                                                    (short)0, acc, false, false);
    }
    const int col   = tn * 16 + l15;
    const float add = bv[col];
    const int rbase = tm * 16 + (lane >> 4) * 8;       // C/D layout: M = v (+8)
#pragma unroll
    for (int v = 0; v < 8; ++v)
        base[(rbase + v) * NQD + col] = acc[v] + add;
#else
    int idx = blockIdx.x * blockDim.x + threadIdx.x;
    for (int o = idx; o < NBD * NQD; o += gridDim.x * blockDim.x) {
        int b = o >> 8, q = o & 255;
        float s = bv[q];
        for (int k = 0; k < NXD; ++k) s = fmaf(xi[b * NXD + k], C1[q * NXD + k], s);
        for (int k = 0; k < NUD; ++k) s = fmaf(u[b * NUD + k], D12[q * NUD + k], s);
        base[o] = s;
    }
#endif
}

// ---- K5: sequential tanh recurrence; 32 batch rows / WGP, LDS tile ----
#define TB 32
__global__ __launch_bounds__(256) void k_rec(const float* __restrict__ base,
                                             const float* __restrict__ D11,
                                             float* __restrict__ w) {
    __shared__ float wt[NQD][TB];     // transposed: column-major dot is conflict-free
    __shared__ float red[256];
    __shared__ float drow[NQD];
    const int tid = threadIdx.x;
    const int r   = tid & 31;         // batch row in tile (lane)
    const int q8  = tid >> 5;         // 8-way K split (wave id)
    const int row0 = blockIdx.x * TB;
    for (int i = 0; i < NQD; ++i) {
        if (tid < i) drow[tid] = D11[i * NQD + tid];
        __syncthreads();
        float partial = 0.f;
        for (int j = q8; j < i; j += 8)
            partial = fmaf(wt[j][r], drow[j], partial);
        red[tid] = partial;
        __syncthreads();
        if (tid < TB) {
            float s = base[(row0 + tid) * NQD + i];
#pragma unroll
            for (int kk = 0; kk < 8; ++kk) s += red[tid + kk * TB];
            wt[i][tid] = tanhf(s);
        }
        __syncthreads();
    }
    for (int idx = tid; idx < TB * NQD; idx += 256) {   // coalesced writeback
        int rr = idx >> 8;
        int cc = idx & 255;
        w[(row0 + rr) * NQD + cc] = wt[cc][rr];
    }
}

// ---- K6: xi_dot = xi@A^T + w@B1^T + u@B2^T + bx  (WMMA f32, K=640) ----
__global__ __launch_bounds__(256) void k_final(const float* __restrict__ xi,
                                               const float* __restrict__ w,
                                               const float* __restrict__ u,
                                               const float* __restrict__ Paug,
                                               const float* __restrict__ B2,
                                               const float* __restrict__ bx,
                                               float* __restrict__ out) {
#if defined(__gfx1250__) && __has_builtin(__builtin_amdgcn_wmma_f32_16x16x4_f32)
    const int lane = threadIdx.x & 31;
    const int wave = threadIdx.x >> 5;
    const int tile = blockIdx.x * 8 + wave;
    const int tm = tile >> 4;
    const int tn = tile & 15;
    const int l15 = lane & 15;
    const int kh  = (lane >> 4) * 2;
    const float* aX  = xi + (tm * 16 + l15) * NXD + kh;
    const float* aW  = w  + (tm * 16 + l15) * NQD + kh;
    const float* aU  = u  + (tm * 16 + l15) * NUD + kh;
    const float* bA  = Paug + (tn * 16 + l15) * AUGW + 256 + kh;  // A[n][k]
    const float* bB1 = Paug + (tn * 16 + l15) * AUGW + 512 + kh;  // B1[n][k]
    const float* bB2 = B2   + (tn * 16 + l15) * NUD + kh;
    v8f acc = {};
    for (int k0 = 0; k0 < NXD; k0 += 4) {
        v2f a = *(const v2f*)(aX + k0);
        v2f b = *(const v2f*)(bA + k0);
        acc = __builtin_amdgcn_wmma_f32_16x16x4_f32(false, a, false, b,
                                                    (short)0, acc, false, false);
    }
    for (int k0 = 0; k0 < NQD; k0 += 4) {
        v2f a = *(const v2f*)(aW + k0);
        v2f b = *(const v2f*)(bB1 + k0);
        acc = __builtin_amdgcn_wmma_f32_16x16x4_f32(false, a, false, b,
                                                    (short)0, acc, false, false);
    }
    for (int k0 = 0; k0 < NUD; k0 += 4) {
        v2f a = *(const v2f*)(aU + k0);
        v2f b = *(const v2f*)(bB2 + k0);
        acc = __builtin_amdgcn_wmma_f32_16x16x4_f32(false, a, false, b,
                                                    (short)0, acc, false, false);
    }
    const int col   = tn * 16 + l15;
    const float add = bx[col];
    const int rbase = tm * 16 + (lane >> 4) * 8;
#pragma unroll
    for (int v = 0; v < 8; ++v)
        out[(rbase + v) * NXD + col] = acc[v] + add;
#else
    int idx = blockIdx.x * blockDim.x + threadIdx.x;
    for (int o = idx; o < NBD * NXD; o += gridDim.x * blockDim.x) {
        int b = o >> 8, n = o & 255;
        float s = bx[n];
        for (int k = 0; k < NXD; ++k) s = fmaf(xi[b * NXD + k], Paug[n * AUGW + 256 + k], s);
        for (int k = 0; k < NQD; ++k) s = fmaf(w[b * NQD + k],  Paug[n * AUGW + 512 + k], s);
        for (int k = 0; k < NUD; ++k) s = fmaf(u[b * NUD + k],  B2[n * NUD + k], s);
        out[o] = s;
    }
#endif
}

extern "C" void kernel_launch(void* const* d_in, const int* in_sizes, int n_in,
                              void* d_out, int out_size, void* d_ws, size_t ws_size,
                              hipStream_t stream) {
    (void)in_sizes; (void)n_in; (void)out_size; (void)ws_size;
    const float* xi    = (const float*)d_in[1];
    const float* u     = (const float*)d_in[2];
    const float* Pstar = (const float*)d_in[3];
    const float* Chi   = (const float*)d_in[4];
    const float* Y1    = (const float*)d_in[5];
    const float* B2    = (const float*)d_in[6];
    const float* D12   = (const float*)d_in[7];
    const float* X     = (const float*)d_in[8];
    const float* bx    = (const float*)d_in[9];
    const float* bv    = (const float*)d_in[10];
    float* out = (float*)d_out;

    float* ws   = (float*)d_ws;                 // ~18.3 MB of f32 scratch
    float* H    = ws;                           // 512*512      = 262144
    float* Paug = ws + 262144;                  // 256*768      = 196608
    float* D11  = ws + 458752;                  // 256*256      =  65536
    float* C1   = ws + 524288;                  // 256*256      =  65536
    float* base = ws + 589824;                  // 8192*256     = 2097152
    float* w    = ws + 2686976;                 // 8192*256     = 2097152

    k_H    <<<1024, 256, 0, stream>>>(X, H);
    k_prep <<< 256, 256, 0, stream>>>(H, Pstar, Y1, Chi, Paug, D11, C1);
    k_gj   <<<   1, 1024, 0, stream>>>(Paug);
    k_base <<<1024, 256, 0, stream>>>(xi, u, C1, D12, bv, base);
    k_rec  <<< 256, 256, 0, stream>>>(base, D11, w);
    k_final<<<1024, 256, 0, stream>>>(xi, w, u, Paug, B2, bx, out);
}